// Attention_87582973100555
// MI455X (gfx1250) — compile-verified
//
#include <hip/hip_runtime.h>
#include <math.h>

// Problem constants (from reference)
constexpr int S = 512;   // encoder length
constexpr int T = 256;   // decoder length
constexpr int B = 8;     // batch
constexpr int H = 128;   // hidden

typedef __attribute__((ext_vector_type(2))) float v2f;
typedef __attribute__((ext_vector_type(8))) float v8f;

// ---------------------------------------------------------------------------
// Kernel 1: fp32 projection GEMM  Y[rows,H] = X[rows,H] * W[H,H]
// One wave computes one 16x16 C tile via V_WMMA_F32_16X16X4_F32, K-loop of 32.
// Block = (32, 8): threadIdx.y selects the N tile (H/16 = 8 tiles).
// Grid.x = rows/16 M tiles. rows (4096, 2048) are multiples of 16, so every
// wave is full (EXEC all ones, as WMMA requires).
// ---------------------------------------------------------------------------
__global__ __launch_bounds__(256) void proj_wmma(const float* __restrict__ X,
                                                 const float* __restrict__ W,
                                                 float* __restrict__ Y) {
    const int lane  = threadIdx.x;        // 0..31
    const int tileN = threadIdx.y;        // 0..7
    const int tileM = blockIdx.x;

    const int m    = lane & 15;           // row within A tile, column within D tile
    const int half = lane >> 4;           // 0: K=0,1   1: K=2,3
    const int gm   = tileM * 16 + m;      // global A row
    const int n    = tileN * 16 + m;      // global B/D column (lane&15)

    const float* __restrict__ arow = X + gm * H;

    v8f acc = {};
    #pragma unroll 4
    for (int k = 0; k < H; k += 4) {
        const int k0 = k + 2 * half;
        v2f a, b;
        // A 16x4 fragment: VGPR0 = K even-of-pair, VGPR1 = K odd-of-pair
        a.x = arow[k0];
        a.y = arow[k0 + 1];
        // B 4x16 fragment (rows striped across lanes, symmetric layout)
        b.x = W[k0 * H + n];
        b.y = W[(k0 + 1) * H + n];
        acc = __builtin_amdgcn_wmma_f32_16x16x4_f32(
            /*neg_a=*/false, a, /*neg_b=*/false, b,
            /*c_mod=*/(short)0, acc, /*reuse_a=*/false, /*reuse_b=*/false);
    }

    // D 16x16 f32: VGPR i holds M = i + 8*half, N = lane&15
    float* __restrict__ yt = Y + (tileM * 16) * H + tileN * 16 + m;
    #pragma unroll
    for (int i = 0; i < 8; ++i) {
        yt[(i + 8 * half) * H] = acc[i];
    }
}

// ---------------------------------------------------------------------------
// Kernel 2: additive attention scores + masked softmax.
// One block per (t, b). 256 threads = 8 wave32.
// Each wave computes one score s at a time: lane holds 4 h-values (coalesced
// float4 load of the L2-resident enc projection), 4x tanh+FMA against the
// LDS-cached dec row and v, then a 5-step wave32 shuffle reduction.
// Then a block-tree softmax over the 512 scores with the enc-length mask,
// and the dec-length mask applied at the store.
// ---------------------------------------------------------------------------
__global__ __launch_bounds__(256) void bahdanau_scores(
        const float* __restrict__ encP,   // (S*B, H) projected encoder
        const float* __restrict__ decP,   // (T*B, H) projected decoder
        const float* __restrict__ v,      // (H)
        const int* __restrict__ enc_lens, // (B)
        const int* __restrict__ dec_lens, // (B)
        float* __restrict__ out) {        // (T, B, S)
    const int t = blockIdx.x;
    const int b = blockIdx.y;

    __shared__ float dec_sh[H];
    __shared__ float vsh[H];
    __shared__ float sc[S];
    __shared__ float red[256];

    const int tid = threadIdx.x;
    if (tid < H) {
        dec_sh[tid] = decP[(t * B + b) * H + tid];
        vsh[tid]    = v[tid];
    }
    __syncthreads();

    const int wave = tid >> 5;
    const int lane = tid & 31;
    const int h0   = lane * 4;
    const float d0 = dec_sh[h0 + 0], d1 = dec_sh[h0 + 1];
    const float d2 = dec_sh[h0 + 2], d3 = dec_sh[h0 + 3];
    const float w0 = vsh[h0 + 0], w1 = vsh[h0 + 1];
    const float w2 = vsh[h0 + 2], w3 = vsh[h0 + 3];

    for (int s = wave; s < S; s += 8) {
        const float4 e = *(const float4*)(encP + (s * B + b) * H + h0);
        float p = w0 * tanhf(d0 + e.x)
                + w1 * tanhf(d1 + e.y)
                + w2 * tanhf(d2 + e.z)
                + w3 * tanhf(d3 + e.w);
        #pragma unroll
        for (int off = 16; off > 0; off >>= 1)
            p += __shfl_xor(p, off, 32);
        if (lane == 0) sc[s] = p;
    }
    __syncthreads();

    const int elen = enc_lens[b];

    // masked max over s (each thread owns s = tid and s = tid + 256)
    const float m0 = (tid < elen)       ? sc[tid]       : -INFINITY;
    const float m1 = (tid + 256 < elen) ? sc[tid + 256] : -INFINITY;
    red[tid] = fmaxf(m0, m1);
    __syncthreads();
    #pragma unroll
    for (int st = 128; st > 0; st >>= 1) {
        if (tid < st) red[tid] = fmaxf(red[tid], red[tid + st]);
        __syncthreads();
    }
    const float mx = red[0];
    __syncthreads();

    // masked exp + sum
    const float e0 = (tid < elen)       ? __expf(sc[tid] - mx)       : 0.0f;
    const float e1 = (tid + 256 < elen) ? __expf(sc[tid + 256] - mx) : 0.0f;
    red[tid] = e0 + e1;
    __syncthreads();
    #pragma unroll
    for (int st = 128; st > 0; st >>= 1) {
        if (tid < st) red[tid] += red[tid + st];
        __syncthreads();
    }
    const float inv = 1.0f / red[0];
    const float dm  = (t < dec_lens[b]) ? 1.0f : 0.0f;

    float* __restrict__ o = out + (t * B + b) * S;
    o[tid]       = e0 * inv * dm;
    o[tid + 256] = e1 * inv * dm;
}

// ---------------------------------------------------------------------------
extern "C" void kernel_launch(void* const* d_in, const int* in_sizes, int n_in,
                              void* d_out, int out_size, void* d_ws, size_t ws_size,
                              hipStream_t stream) {
    const float* enc      = (const float*)d_in[0];   // (S,B,H)
    const float* dec      = (const float*)d_in[1];   // (T,B,H)
    const float* W1       = (const float*)d_in[2];   // (H,H)
    const float* W2       = (const float*)d_in[3];   // (H,H)
    const float* v        = (const float*)d_in[4];   // (H)
    const int*   enc_lens = (const int*)d_in[5];     // (B)
    const int*   dec_lens = (const int*)d_in[6];     // (B)
    float*       out      = (float*)d_out;           // (T,B,S)

    float* encP = (float*)d_ws;                 // S*B*H floats (2 MB)
    float* decP = encP + (size_t)S * B * H;     // T*B*H floats (1 MB)

    dim3 blk(32, 8);
    proj_wmma<<<(S * B) / 16, blk, 0, stream>>>(enc, W1, encP);
    proj_wmma<<<(T * B) / 16, blk, 0, stream>>>(dec, W2, decP);

    bahdanau_scores<<<dim3(T, B), 256, 0, stream>>>(encP, decP, v,
                                                    enc_lens, dec_lens, out);
}